// SSDLoss_20246475833960
// MI455X (gfx1250) — compile-verified
//
#include <hip/hip_runtime.h>
#include <hip/hip_bf16.h>

#define NUM_CLASSES 81
#define BB 64
#define NN 8732
#define BN (BB * NN)              // 558848
#define ROWS_PER_TILE 128
#define TILE_BYTES (ROWS_PER_TILE * NUM_CLASSES * 4)   // 41472
#define NTILES (BN / ROWS_PER_TILE)                    // 4366
#define SENT 0xFFFFFFFFu

typedef __attribute__((ext_vector_type(2))) float v2f;
typedef __attribute__((ext_vector_type(8))) float v8f;

struct Ctl {
    unsigned pos_cnt;      // 0
    unsigned neg_cnt;      // 1
    float    pos_loss;     // 2
    float    loc_sl1;      // 3
    unsigned num_neg;      // 4  min(neg, 3*pos)
    unsigned bin_hi;       // 5  top-16 threshold bin (SENT if none)
    unsigned need2;        // 6  remaining to pick inside bin_hi
    unsigned key_thresh;   // 7  exact 32-bit key threshold
    unsigned need3;        // 8  count selected at exactly key_thresh
    float    neg_sel_sum;  // 9  sum of neg losses with key > key_thresh
    unsigned pad[6];       // pad to 16 dwords
};

// order-preserving float -> uint (larger float => larger key)
__device__ __forceinline__ unsigned fkey(float f) {
    unsigned u = __float_as_uint(f);
    return (u & 0x80000000u) ? ~u : (u | 0x80000000u);
}
__device__ __forceinline__ float keyval(unsigned k) {
    unsigned u = (k & 0x80000000u) ? (k ^ 0x80000000u) : ~k;
    return __uint_as_float(u);
}

__device__ __forceinline__ void async_load_b128(unsigned lds_off, const void* g) {
    // GLOBAL_LOAD_ASYNC_TO_LDS_B128: VDST = LDS byte offset VGPR, VADDR = 64-bit addr, SADDR=off
    asm volatile("global_load_async_to_lds_b128 %0, %1, off"
                 :: "v"(lds_off), "v"(g) : "memory");
}
__device__ __forceinline__ void wait_async0() {
#if __has_builtin(__builtin_amdgcn_s_wait_asynccnt)
    __builtin_amdgcn_s_wait_asynccnt(0);
#else
    asm volatile("s_wait_asynccnt 0x0" ::: "memory");
#endif
}

__global__ void zero_kernel(unsigned* p, int n) {
    int i = blockIdx.x * 256 + threadIdx.x;
    if (i < n) p[i] = 0u;
}

// One block = one 128-row tile. 8 waves, each wave handles 16 rows.
__global__ __launch_bounds__(256) void rowloss_kernel(
    const float* __restrict__ conf, const int* __restrict__ tgt,
    const float* __restrict__ locp, const float* __restrict__ loct,
    float* __restrict__ loss_out, unsigned* __restrict__ hist1, Ctl* ctl)
{
    __shared__ float s_conf[ROWS_PER_TILE * NUM_CLASSES];   // 41472 B
    __shared__ float s_sums[8][16];
    __shared__ unsigned s_pos, s_neg;
    __shared__ float s_ploss, s_sl1;

    const int tid = threadIdx.x;
    if (tid == 0) { s_pos = 0u; s_neg = 0u; s_ploss = 0.f; s_sl1 = 0.f; }

    const int tile = blockIdx.x;
    const int row0 = tile * ROWS_PER_TILE;
    const char* gsrc = (const char*)(conf + (size_t)row0 * NUM_CLASSES);

    // ---- async stage: contiguous 41472B tile -> LDS (2592 x b128) ----
    unsigned lds_base = (unsigned)(unsigned long long)&s_conf[0];
    for (int ofs = tid * 16; ofs < TILE_BYTES; ofs += 256 * 16) {
        async_load_b128(lds_base + (unsigned)ofs, gsrc + ofs);
    }
    wait_async0();
    __syncthreads();

    const int wave = tid >> 5;
    const int lane = tid & 31;
    const int r    = lane & 15;
    const int half = lane >> 4;
    const int lrow = wave * 16 + r;                  // tile-local row
    const float* srow = &s_conf[lrow * NUM_CLASSES];

    // ---- row max (stable logsumexp): two lanes per row, then xor-16 combine ----
    float m = -3.402823466e38f;
    if (half == 0) { for (int c = 0;  c < 40; ++c) m = fmaxf(m, srow[c]); }
    else           { for (int c = 40; c < 81; ++c) m = fmaxf(m, srow[c]); }
    m = fmaxf(m, __shfl_xor(m, 16, 32));

    // ---- sum of exp via WMMA F32 16x16x4: A = exp tile, B = ones ----
    v8f acc = {};
    v2f bones; bones.x = 1.0f; bones.y = 1.0f;
    for (int kc = 0; kc < 20; ++kc) {
        const int c0 = kc * 4 + half * 2;            // A layout: lanes0-15 K={0,1}, lanes16-31 K={2,3}
        v2f a;
        a.x = __expf(srow[c0]     - m);
        a.y = __expf(srow[c0 + 1] - m);
        acc = __builtin_amdgcn_wmma_f32_16x16x4_f32(
            false, a, false, bones, (short)0, acc, false, false);
    }
    float texp = 0.f;
    if (half == 0) texp = __expf(srow[80] - m);      // tail class 80

    // D layout: lanes 0-15 hold rowsums M=0..7 in acc[0..7]; lanes 16-31 hold M=8..15
    if (lane == 0) {
#pragma unroll
        for (int j = 0; j < 8; ++j) s_sums[wave][j] = acc[j];
    }
    if (lane == 16) {
#pragma unroll
        for (int j = 0; j < 8; ++j) s_sums[wave][8 + j] = acc[j];
    }
    __syncthreads();

    // ---- per-row loss, counters, histogram, smooth-L1 (lanes 0..15 own rows) ----
    if (half == 0) {
        const float rowsum = s_sums[wave][r] + texp;
        const int grow = row0 + lrow;
        const int t = tgt[grow];
        const float xt = s_conf[lrow * NUM_CLASSES + t];
        const float loss = m + __logf(rowsum) - xt;
        loss_out[grow] = loss;
        if (t == 0) {
            atomicAdd(&s_neg, 1u);
            atomicAdd(&hist1[fkey(loss) >> 16], 1u);
        } else {
            atomicAdd(&s_pos, 1u);
            atomicAdd(&s_ploss, loss);
            const float* lp = locp + (size_t)grow * 4;
            const float* lt = loct + (size_t)grow * 4;
            float sl = 0.f;
#pragma unroll
            for (int i = 0; i < 4; ++i) {
                float d = fabsf(lp[i] - lt[i]);
                sl += (d < 1.0f) ? 0.5f * d * d : d - 0.5f;
            }
            atomicAdd(&s_sl1, sl);
        }
    }
    __syncthreads();
    if (tid == 0) {
        atomicAdd(&ctl->pos_cnt, s_pos);
        atomicAdd(&ctl->neg_cnt, s_neg);
        atomicAdd(&ctl->pos_loss, s_ploss);
        atomicAdd(&ctl->loc_sl1, s_sl1);
    }
}

// Radix-select level over a 65536-bin histogram; phase 0 -> bin_hi, phase 1 -> key_thresh.
__global__ __launch_bounds__(256) void select_kernel(const unsigned* __restrict__ hist,
                                                     Ctl* ctl, int phase)
{
    __shared__ unsigned s_target;
    __shared__ unsigned s_csum[256];
    __shared__ unsigned s_above[256];
    const int t = threadIdx.x;
    if (t == 0) {
        unsigned target;
        if (phase == 0) {
            unsigned np = ctl->pos_cnt, nn = ctl->neg_cnt;
            unsigned lim = (np > 0x55555555u) ? SENT : np * 3u;
            target = (nn < lim) ? nn : lim;
            ctl->num_neg = target;
            if (target == 0) {
                ctl->bin_hi = SENT; ctl->need2 = 0;
                ctl->key_thresh = SENT; ctl->need3 = 0;
            }
        } else {
            target = (ctl->bin_hi == SENT) ? 0u : ctl->need2;
        }
        s_target = target;
    }
    __syncthreads();
    const unsigned target = s_target;
    if (target == 0) return;

    const int lo = t * 256;
    unsigned cs = 0;
    for (int j = 0; j < 256; ++j) cs += hist[lo + j];
    s_csum[t] = cs;
    __syncthreads();
    if (t == 0) {
        unsigned run = 0;
        for (int u = 255; u >= 0; --u) { s_above[u] = run; run += s_csum[u]; }
    }
    __syncthreads();
    const unsigned above = s_above[t];
    if (above < target && above + s_csum[t] >= target) {
        unsigned cum = above;
        for (int b = lo + 255; b >= lo; --b) {
            unsigned h = hist[b];
            if (cum + h >= target) {
                if (phase == 0) { ctl->bin_hi = (unsigned)b; ctl->need2 = target - cum; }
                else {
                    ctl->key_thresh = (ctl->bin_hi << 16) | (unsigned)b;
                    ctl->need3 = target - cum;
                }
                break;
            }
            cum += h;
        }
    }
}

__global__ void hist2_fill(const float* __restrict__ loss, const int* __restrict__ tgt,
                           unsigned* __restrict__ hist2, const Ctl* ctl)
{
    const unsigned bin_hi = ctl->bin_hi;
    if (bin_hi == SENT) return;
    const int idx = blockIdx.x * 256 + threadIdx.x;
    if (idx >= BN) return;
    if (tgt[idx] != 0) return;
    const unsigned k = fkey(loss[idx]);
    if ((k >> 16) == bin_hi) atomicAdd(&hist2[k & 0xFFFFu], 1u);
}

__global__ __launch_bounds__(256) void final_sum(const float* __restrict__ loss,
                                                 const int* __restrict__ tgt, Ctl* ctl)
{
    __shared__ float s_sum;
    if (threadIdx.x == 0) s_sum = 0.f;
    __syncthreads();
    const unsigned kt = ctl->key_thresh;
    const int idx = blockIdx.x * 256 + threadIdx.x;
    float v = 0.f;
    if (idx < BN && tgt[idx] == 0) {
        const float l = loss[idx];
        if (fkey(l) > kt) v = l;         // strictly-above; ties handled via need3
    }
#pragma unroll
    for (int o = 16; o > 0; o >>= 1) v += __shfl_xor(v, o, 32);
    if ((threadIdx.x & 31) == 0 && v != 0.f) atomicAdd(&s_sum, v);
    __syncthreads();
    if (threadIdx.x == 0 && s_sum != 0.f) atomicAdd(&ctl->neg_sel_sum, s_sum);
}

__global__ void finalize_kernel(const Ctl* ctl, float* out) {
    const unsigned np = ctl->pos_cnt;
    const unsigned nn = ctl->num_neg;
    float conf_sum = ctl->pos_loss + ctl->neg_sel_sum;
    if (nn > 0 && ctl->key_thresh != SENT && ctl->need3 > 0)
        conf_sum += (float)ctl->need3 * keyval(ctl->key_thresh);
    const unsigned cnt = np + nn;
    const float conf_loss = cnt ? conf_sum / (float)cnt : 0.f;
    const float loc_loss  = np ? ctl->loc_sl1 / (4.0f * (float)np) : 0.f;
    out[0] = conf_loss + loc_loss;
    out[1] = conf_loss;
    out[2] = loc_loss;
}

extern "C" void kernel_launch(void* const* d_in, const int* in_sizes, int n_in,
                              void* d_out, int out_size, void* d_ws, size_t ws_size,
                              hipStream_t stream) {
    const float* loc_pred    = (const float*)d_in[0];
    const float* conf_pred   = (const float*)d_in[1];
    const float* loc_target  = (const float*)d_in[2];
    const int*   conf_target = (const int*)d_in[3];
    (void)d_in; (void)in_sizes; (void)n_in; (void)out_size; (void)ws_size;
    float* out = (float*)d_out;

    char* ws = (char*)d_ws;
    float*    loss  = (float*)ws;                         // BN floats
    unsigned* hist1 = (unsigned*)(ws + (size_t)BN * 4);   // 65536
    unsigned* hist2 = hist1 + 65536;                      // 65536
    Ctl*      ctl   = (Ctl*)(hist2 + 65536);

    const int zn = 2 * 65536 + (int)(sizeof(Ctl) / 4);
    zero_kernel<<<(zn + 255) / 256, 256, 0, stream>>>(hist1, zn);
    rowloss_kernel<<<NTILES, 256, 0, stream>>>(conf_pred, conf_target,
                                               loc_pred, loc_target,
                                               loss, hist1, ctl);
    select_kernel<<<1, 256, 0, stream>>>(hist1, ctl, 0);
    hist2_fill<<<(BN + 255) / 256, 256, 0, stream>>>(loss, conf_target, hist2, ctl);
    select_kernel<<<1, 256, 0, stream>>>(hist2, ctl, 1);
    final_sum<<<(BN + 255) / 256, 256, 0, stream>>>(loss, conf_target, ctl);
    finalize_kernel<<<1, 1, 0, stream>>>(ctl, out);
}